// ARMLoss_55147380080909
// MI455X (gfx1250) — compile-verified
//
#include <hip/hip_runtime.h>
#include <stdint.h>

#define TPB 256

typedef __attribute__((ext_vector_type(2))) float v2f;
typedef __attribute__((ext_vector_type(8))) float v8f;

// ---------------------------------------------------------------------------
// Wave-wide (wave32) sum reduction through the matrix pipe:
// A (16x4 f32) holds one partial per lane slot (a[1]=0), B = all-ones.
// D[m][n] = rowsum_m for every n, so summing the 8 C/D VGPRs per lane gives
// lanes 0-15 the m=0..7 half-sum and lanes 16-31 the m=8..15 half-sum;
// one xor-16 shuffle completes the full wave sum on every lane.
// Requires EXEC all-ones: only call from fully-converged code.
// ---------------------------------------------------------------------------
__device__ __forceinline__ float wave_sum_wmma(float x) {
  v2f a;  a[0] = x;    a[1] = 0.0f;
  v2f bm; bm[0] = 1.0f; bm[1] = 1.0f;
  v8f c = {};
  c = __builtin_amdgcn_wmma_f32_16x16x4_f32(false, a, false, bm,
                                            (short)0, c, false, false);
  float q = c[0] + c[1] + c[2] + c[3] + c[4] + c[5] + c[6] + c[7];
  q += __shfl_xor(q, 16, 32);
  return q;
}

// ---------------------------------------------------------------------------
// K1: zero the per-(b,t) packed best-prior keys.
// ---------------------------------------------------------------------------
__global__ void k_init(unsigned long long* __restrict__ bestPrior, int n) {
  int i = blockIdx.x * blockDim.x + threadIdx.x;
  if (i < n) bestPrior[i] = 0ull;
}

// ---------------------------------------------------------------------------
// K2: matching. One block = 256 priors of one batch row.
// Truth tile staged into LDS via gfx1250 async global->LDS loads.
// code[b*P+p] = (best_truth_idx << 1) | (best_iou >= 0.5)
// bestPrior[b*T+t] = atomicMax over packed (iou_bits<<32)|(0xFFFFFFFF-p)
// (IoU >= 0 so float bits are monotonic as u32; ~p breaks ties toward the
//  lowest prior index, matching jnp.argmax first-occurrence semantics.)
// ---------------------------------------------------------------------------
__global__ void k_match(const float* __restrict__ priors,
                        const float* __restrict__ targets,
                        unsigned* __restrict__ code,
                        unsigned long long* __restrict__ bestPrior,
                        int P, int T) {
  const int b   = blockIdx.y;
  const int tid = threadIdx.x;
  const int p   = blockIdx.x * TPB + tid;

  __shared__ float st[512];                   // T*5 floats (T <= 102)
  __shared__ unsigned long long sbest[64];    // per-truth packed max (T <= 64)

  const int nT5 = T * 5;
  const unsigned long long tgt_base =
      (unsigned long long)(uintptr_t)(targets + (size_t)b * nT5);

  // Async copy of the truth tile into LDS (ASYNCcnt-tracked).
  for (int i = tid; i < nT5; i += TPB) {
    unsigned lds_addr = (unsigned)(uintptr_t)(&st[i]);
    unsigned voff     = (unsigned)(i * 4);
    asm volatile("global_load_async_to_lds_b32 %0, %1, %2 offset:0"
                 :: "v"(lds_addr), "v"(voff), "s"(tgt_base) : "memory");
  }
  if (tid < T) sbest[tid] = 0ull;
  asm volatile("s_wait_asynccnt 0x0" ::: "memory");
  __syncthreads();

  if (p < P) {
    const float pcx = priors[(size_t)p * 4 + 0];
    const float pcy = priors[(size_t)p * 4 + 1];
    const float pw  = priors[(size_t)p * 4 + 2];
    const float ph  = priors[(size_t)p * 4 + 3];
    const float px0 = pcx - pw * 0.5f, py0 = pcy - ph * 0.5f;
    const float px1 = pcx + pw * 0.5f, py1 = pcy + ph * 0.5f;
    const float areap = (px1 - px0) * (py1 - py0);

    float best = -1.0f;
    int   bidx = 0;
    const unsigned long long plo = (unsigned long long)(0xFFFFFFFFu - (unsigned)p);
    for (int t = 0; t < T; ++t) {
      const float tx0 = st[t * 5 + 0], ty0 = st[t * 5 + 1];
      const float tx1 = st[t * 5 + 2], ty1 = st[t * 5 + 3];
      float w = fminf(tx1, px1) - fmaxf(tx0, px0); w = fmaxf(w, 0.0f);
      float h = fminf(ty1, py1) - fmaxf(ty0, py0); h = fmaxf(h, 0.0f);
      const float inter = w * h;
      const float areat = (tx1 - tx0) * (ty1 - ty0);
      const float iou   = inter / (areat + areap - inter);
      if (iou > best) { best = iou; bidx = t; }   // strict '>' keeps first max
      const unsigned long long pk =
          ((unsigned long long)__float_as_uint(iou) << 32) | plo;
      atomicMax(&sbest[t], pk);
    }
    code[(size_t)b * P + p] = ((unsigned)bidx << 1) | (best >= 0.5f ? 1u : 0u);
  }
  __syncthreads();
  if (tid < T) atomicMax(&bestPrior[(size_t)b * T + tid], sbest[tid]);
}

// ---------------------------------------------------------------------------
// K3: sequential fixup per batch: for t = 0..T-1 (last wins, matching the
// scatter order of .at[].set), force-match each truth's best prior.
// ---------------------------------------------------------------------------
__global__ void k_fix(const unsigned long long* __restrict__ bestPrior,
                      unsigned* __restrict__ code, int P, int T) {
  const int b = blockIdx.x;
  if (threadIdx.x == 0) {
    for (int t = 0; t < T; ++t) {
      const unsigned long long pk = bestPrior[(size_t)b * T + t];
      const unsigned p = 0xFFFFFFFFu - (unsigned)(pk & 0xFFFFFFFFull);
      code[(size_t)b * P + p] = ((unsigned)t << 1) | 1u;  // iou forced to 2.0 -> pos
    }
  }
}

// ---------------------------------------------------------------------------
// K4: per-prior losses. Streams loc_pred/conf_pred once; writes the
// negative-mining proxy array and per-chunk partials (sl1_pos, ce_pos, npos)
// reduced per-wave through the WMMA pipe, then combined in fixed order.
// ---------------------------------------------------------------------------
__global__ void k_loss(const float* __restrict__ loc_pred,
                       const float* __restrict__ conf_pred,
                       const float* __restrict__ priors,
                       const float* __restrict__ targets,
                       const unsigned* __restrict__ code,
                       float* __restrict__ proxy,
                       float* __restrict__ chunkPart,
                       int P, int T, int chunks) {
  const int b     = blockIdx.y;
  const int chunk = blockIdx.x;
  const int tid   = threadIdx.x;
  const int p     = chunk * TPB + tid;

  __shared__ float st[512];
  __shared__ float lred[(TPB / 32) * 3];

  const int nT5 = T * 5;
  for (int i = tid; i < nT5; i += TPB)
    st[i] = targets[(size_t)b * nT5 + i];
  __syncthreads();

  float sl1 = 0.0f, cep = 0.0f, posf = 0.0f;
  if (p < P) {
    const unsigned cd = code[(size_t)b * P + p];
    const int pos = (int)(cd & 1u);
    const int t   = (int)(cd >> 1);

    const size_t ci = ((size_t)b * P + p) * 2;
    const float x0 = conf_pred[ci + 0], x1 = conf_pred[ci + 1];
    const float m   = fmaxf(x0, x1);
    const float lse = m + log1pf(expf(fminf(x0, x1) - m));
    const float ce  = lse - (pos ? x1 : x0);   // gather class conf_t

    if (pos) {
      posf = 1.0f; cep = ce;
      const float tx0 = st[t * 5 + 0], ty0 = st[t * 5 + 1];
      const float tx1 = st[t * 5 + 2], ty1 = st[t * 5 + 3];
      const float pcx = priors[(size_t)p * 4 + 0];
      const float pcy = priors[(size_t)p * 4 + 1];
      const float pw  = priors[(size_t)p * 4 + 2];
      const float ph  = priors[(size_t)p * 4 + 3];
      float lt[4];
      lt[0] = ((tx0 + tx1) * 0.5f - pcx) / (0.1f * pw);
      lt[1] = ((ty0 + ty1) * 0.5f - pcy) / (0.1f * ph);
      lt[2] = logf((tx1 - tx0) / pw) / 0.2f;
      lt[3] = logf((ty1 - ty0) / ph) / 0.2f;
      const float* lp = &loc_pred[((size_t)b * P + p) * 4];
      #pragma unroll
      for (int j = 0; j < 4; ++j) {
        const float d  = lp[j] - lt[j];
        const float ad = fabsf(d);
        sl1 += (ad < 1.0f) ? 0.5f * d * d : ad - 0.5f;
      }
    }
    proxy[(size_t)b * P + p] = pos ? 0.0f : ce;  // positives pinned to +0
  }
  __syncthreads();

  // fully converged: wave-level WMMA reductions, then fixed-order combine
  const float w0 = wave_sum_wmma(sl1);
  const float w1 = wave_sum_wmma(cep);
  const float w2 = wave_sum_wmma(posf);
  const int wid = tid >> 5, lane = tid & 31;
  if (lane == 0) {
    lred[wid * 3 + 0] = w0; lred[wid * 3 + 1] = w1; lred[wid * 3 + 2] = w2;
  }
  __syncthreads();
  if (tid == 0) {
    float a0 = 0.f, a1 = 0.f, a2 = 0.f;
    for (int w = 0; w < (TPB / 32); ++w) {
      a0 += lred[w * 3 + 0]; a1 += lred[w * 3 + 1]; a2 += lred[w * 3 + 2];
    }
    const size_t o = ((size_t)b * chunks + chunk) * 3;
    chunkPart[o + 0] = a0; chunkPart[o + 1] = a1; chunkPart[o + 2] = a2;
  }
}

// ---------------------------------------------------------------------------
// K5: per batch row: combine chunk partials, then sum of top-num_neg proxy
// values via 4-pass MSB radix select (exact, tie-correct: S_gt + kk * x).
// ---------------------------------------------------------------------------
__global__ void k_batch(const float* __restrict__ proxy,
                        const float* __restrict__ chunkPart,
                        float* __restrict__ batchOut,
                        int P, int chunks) {
  const int b = blockIdx.x, tid = threadIdx.x;
  __shared__ float r0[TPB], r1[TPB], r2[TPB];

  float a0 = 0.f, a1 = 0.f, a2 = 0.f;
  for (int c = tid; c < chunks; c += TPB) {
    const size_t o = ((size_t)b * chunks + c) * 3;
    a0 += chunkPart[o + 0]; a1 += chunkPart[o + 1]; a2 += chunkPart[o + 2];
  }
  r0[tid] = a0; r1[tid] = a1; r2[tid] = a2;
  __syncthreads();
  for (int off = TPB / 2; off > 0; off >>= 1) {
    if (tid < off) { r0[tid] += r0[tid + off]; r1[tid] += r1[tid + off]; r2[tid] += r2[tid + off]; }
    __syncthreads();
  }
  __shared__ float sh_sl1, sh_cep; __shared__ int sh_npos;
  if (tid == 0) { sh_sl1 = r0[0]; sh_cep = r1[0]; sh_npos = (int)(r2[0] + 0.5f); }
  __syncthreads();

  const int npos = sh_npos;
  const int k = min(3 * npos, P - npos);
  float topk = 0.0f;

  if (k > 0) {
    __shared__ unsigned hist[256];
    __shared__ unsigned sh_pref; __shared__ int sh_k;
    const float* row = &proxy[(size_t)b * P];
    unsigned pref = 0; int kk = k;
    for (int level = 3; level >= 0; --level) {
      hist[tid] = 0u;
      __syncthreads();
      const int shift = level * 8;
      const unsigned maskhi = (level == 3) ? 0u : (0xFFFFFFFFu << (shift + 8));
      for (int i = tid; i < P; i += TPB) {
        const unsigned bits = __float_as_uint(row[i]);
        if ((bits & maskhi) == pref) atomicAdd(&hist[(bits >> shift) & 0xFFu], 1u);
      }
      __syncthreads();
      if (tid == 0) {
        int acc = 0, bin = 255;
        for (; bin > 0; --bin) {
          const int h = (int)hist[bin];
          if (acc + h >= kk) break;
          acc += h;
        }
        sh_pref = pref | ((unsigned)bin << shift);
        sh_k    = kk - acc;
      }
      __syncthreads();
      pref = sh_pref; kk = sh_k;
    }
    const float x = __uint_as_float(pref);   // value of the k-th largest
    float s = 0.0f;
    for (int i = tid; i < P; i += TPB) {
      const float v = row[i];
      if (v > x) s += v;
    }
    r0[tid] = s;
    __syncthreads();
    for (int off = TPB / 2; off > 0; off >>= 1) {
      if (tid < off) r0[tid] += r0[tid + off];
      __syncthreads();
    }
    topk = r0[0] + (float)kk * x;            // kk copies of x inside top-k
  }

  if (tid == 0) {
    batchOut[b * 3 + 0] = sh_sl1;
    batchOut[b * 3 + 1] = sh_cep + topk;
    batchOut[b * 3 + 2] = (float)npos;
  }
}

// ---------------------------------------------------------------------------
// K6: final scalars.
// ---------------------------------------------------------------------------
__global__ void k_final(const float* __restrict__ batchOut,
                        float* __restrict__ out, int B) {
  if (blockIdx.x == 0 && threadIdx.x == 0) {
    float ll = 0.f, lc = 0.f, np = 0.f;
    for (int b = 0; b < B; ++b) {
      ll += batchOut[b * 3 + 0];
      lc += batchOut[b * 3 + 1];
      np += batchOut[b * 3 + 2];
    }
    out[0] = ll / np;
    out[1] = lc / np;
  }
}

static inline size_t align16(size_t x) { return (x + 15) & ~(size_t)15; }

extern "C" void kernel_launch(void* const* d_in, const int* in_sizes, int n_in,
                              void* d_out, int out_size, void* d_ws, size_t ws_size,
                              hipStream_t stream) {
  (void)n_in; (void)out_size; (void)ws_size;
  const float* loc_pred  = (const float*)d_in[0];
  const float* conf_pred = (const float*)d_in[1];
  const float* priors    = (const float*)d_in[2];
  const float* targets   = (const float*)d_in[3];

  const int P = in_sizes[2] / 4;                 // priors [P,4]
  const int B = in_sizes[0] / (P * 4);           // loc_pred [B,P,4]
  const int T = in_sizes[3] / (B * 5);           // targets [B,T,5]
  const int chunks = (P + TPB - 1) / TPB;

  char* w = (char*)d_ws;
  unsigned*            code      = (unsigned*)w;            w += align16((size_t)B * P * 4);
  float*               proxy     = (float*)w;               w += align16((size_t)B * P * 4);
  unsigned long long*  bestPrior = (unsigned long long*)w;  w += align16((size_t)B * T * 8);
  float*               chunkPart = (float*)w;               w += align16((size_t)B * chunks * 3 * 4);
  float*               batchOut  = (float*)w;

  k_init<<<(B * T + TPB - 1) / TPB, TPB, 0, stream>>>(bestPrior, B * T);

  dim3 grid(chunks, B);
  k_match<<<grid, TPB, 0, stream>>>(priors, targets, code, bestPrior, P, T);
  k_fix<<<B, 64, 0, stream>>>(bestPrior, code, P, T);
  k_loss<<<grid, TPB, 0, stream>>>(loc_pred, conf_pred, priors, targets,
                                   code, proxy, chunkPart, P, T, chunks);
  k_batch<<<B, TPB, 0, stream>>>(proxy, chunkPart, batchOut, P, chunks);
  k_final<<<1, 64, 0, stream>>>(batchOut, (float*)d_out, B);
}